// MinMaxNorm_34961033790076
// MI455X (gfx1250) — compile-verified
//
#include <hip/hip_runtime.h>

#define NUM_SEG 2048
#define DFEAT   64

typedef float v4f __attribute__((ext_vector_type(4)));
typedef float v2f __attribute__((ext_vector_type(2)));

// ---- order-preserving float <-> u32 encoding (monotonic under unsigned cmp) ----
__device__ __forceinline__ unsigned fenc(float f) {
  unsigned u = __float_as_uint(f);
  return (u & 0x80000000u) ? ~u : (u | 0x80000000u);
}
__device__ __forceinline__ float fdec(unsigned u) {
  return (u & 0x80000000u) ? __uint_as_float(u ^ 0x80000000u)
                           : __uint_as_float(~u);
}

// Low 32 bits of a generic pointer to __shared__ = wave-relative LDS byte offset.
__device__ __forceinline__ unsigned lds_off(const void* p) {
  return (unsigned)(uintptr_t)p;
}

// gfx1250 async global->LDS copy, 16B per lane (tracked by ASYNCcnt).
__device__ __forceinline__ void async_ld_b128(unsigned lds_addr, const float* gaddr) {
  asm volatile("global_load_async_to_lds_b128 %0, %1, off"
               :: "v"(lds_addr), "v"(gaddr) : "memory");
}

// ---- kernel 1: init per-segment accumulators ----
__global__ void seg_init(unsigned* __restrict__ minU, unsigned* __restrict__ maxU) {
  int g = blockIdx.x * blockDim.x + threadIdx.x;
  if (g < NUM_SEG) {
    minU[g] = 0xFFFFFFFFu;  // enc(+inf) <= this
    maxU[g] = 0u;           // enc(-inf) >= this
  }
}

// 16-lane-group butterfly reduce + one atomic flush per group.
__device__ __forceinline__ void flush_group(int segId, float fm, float fM, int col4,
                                            unsigned* __restrict__ minU,
                                            unsigned* __restrict__ maxU) {
  #pragma unroll
  for (int off = 1; off < 16; off <<= 1) {   // masks 1,2,4,8 stay inside the group
    fm = fminf(fm, __shfl_xor(fm, off, 32));
    fM = fmaxf(fM, __shfl_xor(fM, off, 32));
  }
  if (col4 == 0) {
    atomicMin(&minU[segId], fenc(fm));
    atomicMax(&maxU[segId], fenc(fM));
  }
}

__device__ __forceinline__ void accum_row(const v4f v, int s, int& curSeg,
                                          float& curMin, float& curMax, int col4,
                                          unsigned* __restrict__ minU,
                                          unsigned* __restrict__ maxU) {
  const float m = fminf(fminf(v.x, v.y), fminf(v.z, v.w));
  const float M = fmaxf(fmaxf(v.x, v.y), fmaxf(v.z, v.w));
  if (s != curSeg) {                        // group-uniform (rare: ~488 rows/segment)
    if (curSeg >= 0) flush_group(curSeg, curMin, curMax, col4, minU, maxU);
    curSeg = s; curMin = m; curMax = M;
  } else {
    curMin = fminf(curMin, m);
    curMax = fmaxf(curMax, M);
  }
}

// ---- kernel 2: per-segment scalar min/max over rows (seg is sorted) ----
// Block = 256 threads = 8 waves; each wave owns 32 contiguous rows. Per
// iteration lanes 0-15 serve row r, lanes 16-31 row r+1 (one float4/lane ->
// the wave touches 512 contiguous bytes). Full blocks stream through a
// double-buffered async global->LDS pipeline (ASYNCcnt); the tail block uses
// guarded direct b128 loads.
__global__ __launch_bounds__(256) void seg_minmax(
    const float* __restrict__ x, const int* __restrict__ seg,
    unsigned* __restrict__ minU, unsigned* __restrict__ maxU, int N)
{
  __shared__ v4f stage[2][256];             // 8 KB: 2 buffers x 16B/thread

  const int tid  = threadIdx.x;
  const int lane = tid & 31;
  const int wave = tid >> 5;
  const int grp  = lane >> 4;               // 0 or 1: which row of the pair
  const int col4 = lane & 15;               // float4 slot within the 64-wide row
  const int waveBase = blockIdx.x * 256 + wave * 32;

  int   curSeg = -1;
  float curMin =  __builtin_inff();
  float curMax = -__builtin_inff();

  const bool fullBlock = (blockIdx.x * 256 + 256) <= N;

  if (fullBlock) {
    // prologue: fill buffer 0 with iteration 0's row pair
    async_ld_b128(lds_off(&stage[0][tid]),
                  x + (size_t)(waveBase + grp) * DFEAT + col4 * 4);

    #pragma unroll
    for (int it = 0; it < 16; ++it) {
      if (it < 15) {
        // WAR: prior ds_load of the buffer we are about to overwrite must retire
        asm volatile("s_wait_dscnt 0x0" ::: "memory");
        async_ld_b128(lds_off(&stage[(it + 1) & 1][tid]),
                      x + (size_t)(waveBase + 2 * (it + 1) + grp) * DFEAT + col4 * 4);
        asm volatile("s_wait_asynccnt 0x1" ::: "memory");  // current buffer filled
      } else {
        asm volatile("s_wait_asynccnt 0x0" ::: "memory");  // last buffer filled
      }
      const v4f v = stage[it & 1][tid];
      const int row = waveBase + 2 * it + grp;
      accum_row(v, seg[row], curSeg, curMin, curMax, col4, minU, maxU);
    }
  } else {
    #pragma unroll 4
    for (int it = 0; it < 16; ++it) {
      const int row = waveBase + 2 * it + grp;
      if (row < N) {                        // group-uniform predicate
        const v4f v = *reinterpret_cast<const v4f*>(
            x + (size_t)row * DFEAT + col4 * 4);
        accum_row(v, seg[row], curSeg, curMin, curMax, col4, minU, maxU);
      }
    }
  }

  if (curSeg >= 0) flush_group(curSeg, curMin, curMax, col4, minU, maxU);
}

// ---- kernel 3: decode + precompute (mn, 1/(mx-mn+eps)) per segment ----
__global__ void seg_finalize(const unsigned* __restrict__ minU,
                             const unsigned* __restrict__ maxU,
                             v2f* __restrict__ mni)
{
  int g = blockIdx.x * blockDim.x + threadIdx.x;
  if (g < NUM_SEG) {
    const float mn = fdec(minU[g]);
    const float mx = fdec(maxU[g]);
    v2f p;
    p.x = mn;
    p.y = 1.0f / (mx - mn + 1e-6f);
    mni[g] = p;
  }
}

// ---- kernel 4: normalize, one float4 per thread, NT load/store ----
__global__ __launch_bounds__(256) void seg_norm(
    const float* __restrict__ x, const int* __restrict__ seg,
    const v2f* __restrict__ mni, float* __restrict__ out, int n4)
{
  const int i = blockIdx.x * blockDim.x + threadIdx.x;
  if (i >= n4) return;
  const int row = i >> 4;                   // 16 float4s per 64-wide row
  const int g   = seg[row];
  const v2f p   = mni[g];
  const v4f v   = __builtin_nontemporal_load(reinterpret_cast<const v4f*>(x) + i);
  v4f r;
  r.x = (v.x - p.x) * p.y;
  r.y = (v.y - p.x) * p.y;
  r.z = (v.z - p.x) * p.y;
  r.w = (v.w - p.x) * p.y;
  __builtin_nontemporal_store(r, reinterpret_cast<v4f*>(out) + i);
}

extern "C" void kernel_launch(void* const* d_in, const int* in_sizes, int n_in,
                              void* d_out, int out_size, void* d_ws, size_t ws_size,
                              hipStream_t stream) {
  const float* x   = (const float*)d_in[0];
  const int*   seg = (const int*)d_in[1];
  const int    N   = in_sizes[1];            // rows; in_sizes[0] == N*64

  unsigned* minU = (unsigned*)d_ws;          // [2048] u32
  unsigned* maxU = minU + NUM_SEG;           // [2048] u32
  v2f*      mni  = (v2f*)(maxU + NUM_SEG);   // [2048] (mn, inv_denom)

  seg_init<<<(NUM_SEG + 255) / 256, 256, 0, stream>>>(minU, maxU);

  const int rowBlocks = (N + 255) / 256;     // 256 rows per block
  seg_minmax<<<rowBlocks, 256, 0, stream>>>(x, seg, minU, maxU, N);

  seg_finalize<<<(NUM_SEG + 255) / 256, 256, 0, stream>>>(minU, maxU, mni);

  const int n4 = (int)(((size_t)N * DFEAT) / 4);
  seg_norm<<<(n4 + 255) / 256, 256, 0, stream>>>(x, seg, mni, (float*)d_out, n4);
}